// RoIPooling_6880537608987
// MI455X (gfx1250) — compile-verified
//
#include <hip/hip_runtime.h>
#include <stdint.h>
#include <math.h>

// Problem constants from the reference
#define B_ 2
#define C_ 256
#define H_ 64
#define W_ 64
#define SPATIAL_SCALE 1.0f

// One wave32 per (roi, channel). 8 waves per 256-thread block.
// Rows of x are streamed Global->LDS with the CDNA5 async engine
// (ASYNCcnt-tracked, double-buffered), reduced with fmax, then a
// wave32 butterfly reduction produces the per-channel max.
__global__ __launch_bounds__(256) void roi_max_pool_async_kernel(
    const float* __restrict__ x,
    const int*   __restrict__ rois,
    float*       __restrict__ out,
    int K)
{
    // per-wave double buffer: [wave][buf][64 floats] = 4 KB total
    __shared__ float smem[8][2][W_];

    const int lane = threadIdx.x & 31;
    const int wave = threadIdx.x >> 5;
    const int k    = blockIdx.x;
    const int c    = blockIdx.y * 8 + wave;
    if (k >= K) return;

    // ---- ROI decode (exactly mirrors the reference) ----
    const int* r = rois + 5 * k;
    const int bi = r[0];
    const int x1 = __float2int_rn((float)r[1] * SPATIAL_SCALE);
    const int y1 = __float2int_rn((float)r[2] * SPATIAL_SCALE);
    const int x2 = __float2int_rn((float)r[3] * SPATIAL_SCALE);
    const int y2 = __float2int_rn((float)r[4] * SPATIAL_SCALE);
    const int roi_w = max(x2 - x1 + 1, 1);
    const int roi_h = max(y2 - y1 + 1, 1);
    const int hs = min(max(y1, 0), H_);
    const int he = min(max(y1 + roi_h, 0), H_);
    const int ws = min(max(x1, 0), W_);
    const int we = min(max(x1 + roi_w, 0), W_);

    const int  nrows = he - hs;
    const bool empty = (nrows <= 0) || (we <= ws);

    float m = -INFINITY;

    if (!empty) {
        const float* base = x + (((size_t)bi * C_ + c) * H_) * (size_t)W_;

        // LDS byte offset of this wave's staging area (low 32 bits of the
        // generic pointer are the LDS offset on gfx1250).
        const unsigned lbase = (unsigned)(uintptr_t)(&smem[wave][0][0]);
        // lane moves elements 2*lane, 2*lane+1 of a row (B64 = 8 bytes/lane,
        // 32 lanes => one full 256B row per instruction, fully coalesced)
        const unsigned lslot = (unsigned)(lane * 8);

        // loop-invariant column masks for the two LDS slots this lane reads
        const bool ok0 = (lane      >= ws) && (lane      < we);
        const bool ok1 = (lane + 32 >= ws) && (lane + 32 < we);

        // stage(row -> buf): one async Global->LDS B64 per row per wave
        #define STAGE_ROW(ROW, BUF)                                            \
            asm volatile("global_load_async_to_lds_b64 %0, %1, off"            \
                         :: "v"(lbase + (unsigned)((BUF) * (W_ * 4)) + lslot), \
                            "v"(base + (size_t)(ROW) * W_ + 2 * lane)          \
                         : "memory")

        // prologue: fill the double buffer
        STAGE_ROW(hs, 0);
        if (nrows > 1) STAGE_ROW(hs + 1, 1);

        for (int i = 0; i < nrows; ++i) {
            const int buf = i & 1;
            // In-order ASYNCcnt completion: <=1 outstanding means the
            // current buffer's row has landed in LDS.
            if (i + 1 < nrows) {
                asm volatile("s_wait_asynccnt 0x1" ::: "memory");
            } else {
                asm volatile("s_wait_asynccnt 0x0" ::: "memory");
            }

            const float v0 = smem[wave][buf][lane];
            const float v1 = smem[wave][buf][lane + 32];
            m = fmaxf(m, fmaxf(ok0 ? v0 : -INFINITY,
                               ok1 ? v1 : -INFINITY));

            // LDS reads retired -> this buffer may be overwritten by the
            // async engine for row i+2.
            asm volatile("s_wait_dscnt 0x0" ::: "memory");

            if (i + 2 < nrows) STAGE_ROW(hs + i + 2, buf);
        }
        #undef STAGE_ROW

        // wave32 butterfly max-reduction
        #pragma unroll
        for (int off = 16; off > 0; off >>= 1) {
            m = fmaxf(m, __shfl_xor(m, off, 32));
        }
    }

    if (lane == 0) {
        out[(size_t)k * C_ + c] = empty ? 0.0f : m;
    }
}

extern "C" void kernel_launch(void* const* d_in, const int* in_sizes, int n_in,
                              void* d_out, int out_size, void* d_ws, size_t ws_size,
                              hipStream_t stream) {
    const float* x    = (const float*)d_in[0];  // (B, C, H, W) float32
    const int*   rois = (const int*)d_in[1];    // (K, 5) int32
    float*       out  = (float*)d_out;          // (K, C) float32

    const int K = in_sizes[1] / 5;

    dim3 grid((unsigned)K, C_ / 8, 1);
    dim3 block(256, 1, 1);
    roi_max_pool_async_kernel<<<grid, block, 0, stream>>>(x, rois, out, K);
}